// SGConvModel_33071248179459
// MI455X (gfx1250) — compile-verified
//
#include <hip/hip_runtime.h>

// SGConv x2 + linear head for MI455X (gfx1250), wave32.
// deg -> dinv -> pre-scale(+self-loop init) -> scatter (global_atomic_add_f32,
// L2-resident tables) -> WMMA f32 GEMM with async LDS staging; dinv + bias +
// relu fused in the epilogue.

typedef __attribute__((ext_vector_type(2))) float v2f;
typedef __attribute__((ext_vector_type(8))) float v8f;
typedef int v4i __attribute__((vector_size(16)));
typedef __attribute__((address_space(3))) v4i* lds_v4i_t;

#define FEAT 128
#define LDS_STRIDE 132   // 132*4 = 528 B row pitch: 16B-aligned, bank-staggered

#if __has_builtin(__builtin_amdgcn_global_load_async_to_lds_b128)
#define USE_ASYNC_LDS 1
#endif

static __device__ __forceinline__ lds_v4i_t to_lds_v4i(const void* p) {
    // generic LDS address: low 32 bits are the LDS offset (aperture doc §10.2)
    return (lds_v4i_t)(unsigned int)(unsigned long long)p;
}

__global__ void zero_f32(float* __restrict__ p, int n) {
    int i = blockIdx.x * blockDim.x + threadIdx.x;
    if (i < n) p[i] = 0.0f;
}

// deg[dst] += 1 over real edges (self-loop +1 folded into make_dinv)
__global__ void deg_count(const long long* __restrict__ dst,
                          float* __restrict__ deg, int E) {
    int e = blockIdx.x * blockDim.x + threadIdx.x;
    if (e < E) {
        int d = (int)dst[e];
        unsafeAtomicAdd(&deg[d], 1.0f);
    }
}

// dinv = rsqrt(deg + 1); deg+1 >= 1 so the reference's deg>0 guard is moot
__global__ void make_dinv(float* __restrict__ deg, int n) {
    int i = blockIdx.x * blockDim.x + threadIdx.x;
    if (i < n) deg[i] = rsqrtf(deg[i] + 1.0f);
}

// xs = x * dinv[row]; agg initialized to xs (self-loop message, pre-scale)
__global__ void scale_init(const float* __restrict__ x,
                           const float* __restrict__ dinv,
                           float* __restrict__ xs,
                           float* __restrict__ agg, int total) {
    int i = blockIdx.x * blockDim.x + threadIdx.x;
    if (i < total) {
        float v = x[i] * dinv[i >> 7];   // FEAT == 128
        xs[i] = v;
        agg[i] = v;
    }
}

// One wave per edge: lane grabs float4 of xs[src], 4x global_atomic_add_f32
// into agg[dst]. xs/agg tables (20.5 MB) are fully L2-resident.
__global__ void scatter_add(const float* __restrict__ xs,
                            const long long* __restrict__ src,
                            const long long* __restrict__ dst,
                            float* __restrict__ agg, int E) {
    int lane = threadIdx.x & 31;
    int w    = threadIdx.x >> 5;
    int e    = blockIdx.x * 8 + w;
    if (e >= E) return;
    int s = (int)src[e];
    int d = (int)dst[e];
    const float4 v = *(const float4*)(xs + (size_t)s * FEAT + (lane << 2));
    float* out = agg + (size_t)d * FEAT + (lane << 2);
    unsafeAtomicAdd(out + 0, v.x);
    unsafeAtomicAdd(out + 1, v.y);
    unsafeAtomicAdd(out + 2, v.z);
    unsafeAtomicAdd(out + 3, v.w);
}

// C[M x NOUT] = act( dinv[m]? * (A[M x 128] @ W[NOUT x 128]^T) + bias )
// Block = 256 thr = 8 waves; block owns one 16-row M tile (A panel in LDS,
// async-copied); wave w owns N-tile w. Two WMMA accumulation chains for ILP.
template <int NOUT, bool SCALE, bool RELU>
__global__ __launch_bounds__(256) void gemm_wmma(
        const float* __restrict__ A,
        const float* __restrict__ dinv,
        const float* __restrict__ W,
        const float* __restrict__ bias,
        float* __restrict__ C) {
    __shared__ float As[16 * LDS_STRIDE];
    const int m0  = blockIdx.x << 4;
    const int tid = threadIdx.x;

#ifdef USE_ASYNC_LDS
    // 512 x b128 chunks (8 KB), 2 per lane, straight to LDS via ASYNCcnt path
    #pragma unroll
    for (int i = 0; i < 2; ++i) {
        int idx = tid + i * 256;          // chunk id 0..511
        int r   = idx >> 5;               // row 0..15
        int c   = (idx & 31) << 2;        // col 0,4,...,124
        __builtin_amdgcn_global_load_async_to_lds_b128(
            (v4i*)(A + (size_t)(m0 + r) * FEAT + c),
            to_lds_v4i(As + r * LDS_STRIDE + c), 0, 0);
    }
#if __has_builtin(__builtin_amdgcn_s_wait_asynccnt)
    __builtin_amdgcn_s_wait_asynccnt(0);
#else
    asm volatile("s_wait_asynccnt 0x0" ::: "memory");
#endif
#else
    #pragma unroll
    for (int i = 0; i < 8; ++i) {
        int idx = tid + i * 256;
        int r = idx >> 7, c = idx & 127;
        As[r * LDS_STRIDE + c] = A[(size_t)(m0 + r) * FEAT + c];
    }
#endif
    __syncthreads();

    const int lane  = tid & 31;
    const int w     = tid >> 5;
    const int numNT = (NOUT + 15) >> 4;
    if (w >= numNT) return;                 // wave-uniform exit, EXEC stays full

    const int  ncol = (w << 4) + (lane & 15);
    const bool nok  = (NOUT % 16 == 0) ? true : (ncol < NOUT);
    const int  koff = (lane >> 4) << 1;     // 0 or 2

    v8f acc0 = {}, acc1 = {};
    const float* wrow = W + (size_t)(nok ? ncol : 0) * FEAT;
    const float* arow = As + (lane & 15) * LDS_STRIDE + koff;

    #pragma unroll
    for (int kb = 0; kb < FEAT; kb += 8) {
        v2f a0 = *(const v2f*)(arow + kb);
        v2f a1 = *(const v2f*)(arow + kb + 4);
        v2f b0 = *(const v2f*)(wrow + kb + koff);
        v2f b1 = *(const v2f*)(wrow + kb + 4 + koff);
        if (!nok) { b0.x = 0.f; b0.y = 0.f; b1.x = 0.f; b1.y = 0.f; }
        acc0 = __builtin_amdgcn_wmma_f32_16x16x4_f32(
            false, a0, false, b0, (short)0, acc0, false, false);
        acc1 = __builtin_amdgcn_wmma_f32_16x16x4_f32(
            false, a1, false, b1, (short)0, acc1, false, false);
    }

    if (nok) {
        const int  mb = m0 + ((lane >> 4) << 3);  // M = r + 8*(lane>=16)
        const float bv = bias[ncol];
        #pragma unroll
        for (int r = 0; r < 8; ++r) {
            float v = acc0[r] + acc1[r];
            if (SCALE) v *= dinv[mb + r];
            v += bv;
            if (RELU) v = fmaxf(v, 0.0f);
            C[(size_t)(mb + r) * NOUT + ncol] = v;
        }
    }
}

extern "C" void kernel_launch(void* const* d_in, const int* in_sizes, int n_in,
                              void* d_out, int out_size, void* d_ws, size_t ws_size,
                              hipStream_t stream) {
    const float*     x  = (const float*)d_in[0];
    const long long* ei = (const long long*)d_in[1];   // int64 per reference
    const float*     W1 = (const float*)d_in[2];
    const float*     b1 = (const float*)d_in[3];
    const float*     W2 = (const float*)d_in[4];
    const float*     b2 = (const float*)d_in[5];
    const float*     Wl = (const float*)d_in[6];
    const float*     bl = (const float*)d_in[7];
    float*           out = (float*)d_out;

    const int N = in_sizes[0] / FEAT;   // 40000
    const int E = in_sizes[1] / 2;      // 640000
    const int total = N * FEAT;
    const int mt = N / 16;              // 2500 M tiles (N divisible by 16)

    float* deg = (float*)d_ws;                       // N (becomes dinv)
    float* xs  = deg + (size_t)((N + 255) & ~255);   // N*FEAT
    float* agg = xs + (size_t)N * FEAT;              // N*FEAT
    float* h   = agg + (size_t)N * FEAT;             // N*FEAT

    const long long* src = ei;
    const long long* dst = ei + E;

    zero_f32<<<(N + 255) / 256, 256, 0, stream>>>(deg, N);
    deg_count<<<(E + 255) / 256, 256, 0, stream>>>(dst, deg, E);
    make_dinv<<<(N + 255) / 256, 256, 0, stream>>>(deg, N);

    // ---- layer 1 ----
    scale_init<<<(total + 255) / 256, 256, 0, stream>>>(x, deg, xs, agg, total);
    scatter_add<<<(E + 7) / 8, 256, 0, stream>>>(xs, src, dst, agg, E);
    gemm_wmma<FEAT, true, true><<<mt, 256, 0, stream>>>(agg, deg, W1, b1, h);

    // ---- layer 2 ----
    scale_init<<<(total + 255) / 256, 256, 0, stream>>>(h, deg, xs, agg, total);
    scatter_add<<<(E + 7) / 8, 256, 0, stream>>>(xs, src, dst, agg, E);
    gemm_wmma<FEAT, true, true><<<mt, 256, 0, stream>>>(agg, deg, W2, b2, h);

    // ---- linear head (NOUT = 40, partial last tile) ----
    gemm_wmma<40, false, false><<<mt, 256, 0, stream>>>(h, deg, Wl, bl, out);
}